// HungarianMatcher_4466765988574
// MI455X (gfx1250) — compile-verified
//
#include <hip/hip_runtime.h>
#include <hip/hip_bf16.h>

// ---- problem constants (match reference) ----
#define B_      16
#define Q_      2048
#define NCLS_   2
#define DBOX_   6
#define T_      1024
#define COST_CLASS_ 1.0f
#define COST_BBOX_  5.0f

#define TT_PER_WAVE 8                      // t-tiles (of 16 targets) per wave
#define QTILES  (Q_ / 16)                  // 128
#define TGROUPS (T_ / (16 * TT_PER_WAVE))  // 8
#define TOTAL_WAVES (B_ * QTILES * TGROUPS)  // 16384
#define THREADS_PER_BLOCK 256
#define WAVES_PER_BLOCK (THREADS_PER_BLOCK / 32)

typedef __attribute__((ext_vector_type(2))) float v2f;
typedef __attribute__((ext_vector_type(8))) float v8f;

__global__ __launch_bounds__(THREADS_PER_BLOCK)
void hungarian_cost_wmma_kernel(const float* __restrict__ pred_logits,  // (B,Q,2)
                                const float* __restrict__ pred_boxes,   // (B,Q,6)
                                const int*   __restrict__ tgt_labels,   // (T,)
                                const float* __restrict__ tgt_boxes,    // (T,6)
                                float*       __restrict__ out)          // (B,Q,T)
{
    const int lane = threadIdx.x & 31;
    const int wid  = (int)((blockIdx.x * blockDim.x + threadIdx.x) >> 5);

    // wave -> (batch, q-tile, t-group)
    const int b     = wid / (QTILES * TGROUPS);
    const int rem   = wid % (QTILES * TGROUPS);
    const int qtile = rem / TGROUPS;
    const int tg    = rem % TGROUPS;

    const int qbase  = qtile * 16;
    const int tbase0 = tg * (16 * TT_PER_WAVE);

    const int nloc    = lane & 15;   // column (target) within tile / row M for A
    const int half    = lane >> 4;   // lane-half selector
    const int rowbase = half * 8;    // C/D rows handled by this lane

    // ---------------- query-side data, reused across all 8 t-tiles ----------------
    // A-matrix (16x4 f32, ISA layout): lanes 0-15 -> K=0,1 ; lanes 16-31 -> K=2,3 (zero pad)
    v2f amat;
    {
        const float* lg = pred_logits + (size_t)(b * Q_ + qbase + nloc) * NCLS_;
        float l0 = lg[0], l1 = lg[1];
        float mx = fmaxf(l0, l1);
        float e0 = __expf(l0 - mx);
        float e1 = __expf(l1 - mx);
        float inv = 1.0f / (e0 + e1);
        float p0 = e0 * inv;
        float p1 = e1 * inv;
        // fold the minus sign and COST_CLASS into A
        amat.x = (half == 0) ? (-COST_CLASS_ * p0) : 0.0f;
        amat.y = (half == 0) ? (-COST_CLASS_ * p1) : 0.0f;
    }

    // query boxes for the 8 rows this lane owns in C/D
    float qb[8][DBOX_];
#pragma unroll
    for (int i = 0; i < 8; ++i) {
        const float* qp = pred_boxes + (size_t)(b * Q_ + qbase + rowbase + i) * DBOX_;
#pragma unroll
        for (int d = 0; d < DBOX_; ++d) qb[i][d] = qp[d];
    }

    // ---------------- loop over 8 consecutive 16-target tiles ----------------
#pragma unroll
    for (int tt = 0; tt < TT_PER_WAVE; ++tt) {
        const int tbase = tbase0 + tt * 16;
        const int t     = tbase + nloc;

        const int label = tgt_labels[t];

        float tb[DBOX_];
        const float* tp = tgt_boxes + (size_t)t * DBOX_;
#pragma unroll
        for (int d = 0; d < DBOX_; ++d) tb[d] = tp[d];

        // B-matrix (4x16 f32): lanes 0-15 hold rows K=0,1 (one-hot of this
        // column's label); lanes 16-31 hold rows K=2,3 (zero pad).
        v2f bmat;
        bmat.x = (half == 0 && label == 0) ? 1.0f : 0.0f;
        bmat.y = (half == 0 && label == 1) ? 1.0f : 0.0f;

        // accumulator C = COST_BBOX * L1(query_box, target_box), in C/D layout
        v8f cacc;
#pragma unroll
        for (int i = 0; i < 8; ++i) {
            float s = fabsf(qb[i][0] - tb[0]);
            s += fabsf(qb[i][1] - tb[1]);
            s += fabsf(qb[i][2] - tb[2]);
            s += fabsf(qb[i][3] - tb[3]);
            s += fabsf(qb[i][4] - tb[4]);
            s += fabsf(qb[i][5] - tb[5]);
            cacc[i] = COST_BBOX_ * s;
        }

        // D = A x B + C : fuses the class-cost gather with the bbox cost.
        v8f dres = __builtin_amdgcn_wmma_f32_16x16x4_f32(
            /*neg_a=*/false, amat, /*neg_b=*/false, bmat,
            /*c_mod=*/(short)0, cacc, /*reuse_a=*/false, /*reuse_b=*/false);

        // store: lane writes rows rowbase..rowbase+7, column tbase+nloc
        const size_t obase = (size_t)(b * Q_ + qbase + rowbase) * T_ + (size_t)(tbase + nloc);
#pragma unroll
        for (int i = 0; i < 8; ++i) {
            out[obase + (size_t)i * T_] = dres[i];
        }
    }
}

extern "C" void kernel_launch(void* const* d_in, const int* in_sizes, int n_in,
                              void* d_out, int out_size, void* d_ws, size_t ws_size,
                              hipStream_t stream) {
    (void)in_sizes; (void)n_in; (void)out_size; (void)d_ws; (void)ws_size;
    const float* pred_logits = (const float*)d_in[0];
    const float* pred_boxes  = (const float*)d_in[1];
    const int*   tgt_labels  = (const int*)d_in[2];
    const float* tgt_boxes   = (const float*)d_in[3];
    float* out = (float*)d_out;

    const int blocks = TOTAL_WAVES / WAVES_PER_BLOCK;  // 2048
    hungarian_cost_wmma_kernel<<<blocks, THREADS_PER_BLOCK, 0, stream>>>(
        pred_logits, pred_boxes, tgt_labels, tgt_boxes, out);
}